// DGM_c_75806172774562
// MI455X (gfx1250) — compile-verified
//
#include <hip/hip_runtime.h>
#include <math.h>

typedef __attribute__((ext_vector_type(2))) float v2f;
typedef __attribute__((ext_vector_type(8))) float v8f;

constexpr int NN   = 8192;
constexpr int DIN  = 128;
constexpr int DEMB = 4;
constexpr int TOPK = 10;
constexpr int KCHUNKS = 4;                 // K-split for memory-level parallelism

// ---------------------------------------------------------------------------
// K1: Vt = (x @ W)^T, stored [16][NN] (transposed, cols 4..15 of the WMMA
// B-operand zero-padded). Thread = (d, row) so all stores are coalesced.
// Transposed layout makes the GEMM B-fragment a contiguous b64 load.
// ---------------------------------------------------------------------------
__global__ void k_xw(const float* __restrict__ x, const float* __restrict__ W,
                     float* __restrict__ Vt) {
  int t = blockIdx.x * blockDim.x + threadIdx.x;   // 4*NN threads
  int d   = t >> 13;          // 0..3
  int row = t & (NN - 1);
  float acc = 0.f;
  const float* xr = x + (size_t)row * DIN;
#pragma unroll 8
  for (int k = 0; k < DIN; ++k) acc = fmaf(xr[k], W[k * DEMB + d], acc);
  Vt[(size_t)d * NN + row]        = acc;
  Vt[(size_t)(d + 4) * NN + row]  = 0.f;
  Vt[(size_t)(d + 8) * NN + row]  = 0.f;
  Vt[(size_t)(d + 12) * NN + row] = 0.f;
}

// ---------------------------------------------------------------------------
// K2: xe = A @ V via V_WMMA_F32_16X16X4_F32.
// Block = 4 waves = 4 K-chunks of one 16-row block (2048 blocks-waves total ->
// ~11 MB of A in flight, sized for 23.3 TB/s * HBM latency).
// A-frag: lane%16 = M row, lane/16 selects K pair -> nontemporal b64 stream.
// B-frag: lane%16 = N col, contiguous b64 from transposed Vt (L2-resident).
// Partials combined with a deterministic in-block LDS reduction (no atomics).
// ---------------------------------------------------------------------------
__global__ void k_gemm(const float* __restrict__ A, const float* __restrict__ Vt,
                       float* __restrict__ xe) {
  __shared__ float accs[KCHUNKS][64];
  const int lane  = threadIdx.x & 31;
  const int chunk = threadIdx.x >> 5;              // wave id == K chunk
  const int m     = lane & 15;
  const int half  = lane >> 4;
  const int rowBase = blockIdx.x * 16;
  const int kSpan = NN / KCHUNKS;
  const int k0 = chunk * kSpan;
  const int k1 = k0 + kSpan;

  const float* Arow = A  + (size_t)(rowBase + m) * NN;
  const float* Vrow = Vt + (size_t)m * NN;

  v8f c = {};
  for (int k = k0; k < k1; k += 16) {
#pragma unroll
    for (int kk = 0; kk < 16; kk += 4) {
      const int kb = k + kk + 2 * half;
      v2f a = __builtin_nontemporal_load((const v2f*)(Arow + kb)); // stream-once
      v2f b = *(const v2f*)(Vrow + kb);                            // L2-hot
      c = __builtin_amdgcn_wmma_f32_16x16x4_f32(
              /*neg_a=*/false, a, /*neg_b=*/false, b,
              /*c_mod=*/(short)0, c, /*reuse_a=*/false, /*reuse_b=*/false);
    }
  }

  // D layout: vgpr r, lanes 0-15 -> (M=r, N=lane), lanes 16-31 -> (M=r+8).
  if (m < DEMB) {
#pragma unroll
    for (int r = 0; r < 8; ++r)
      accs[chunk][(r + 8 * half) * DEMB + m] = c[r];
  }
  __syncthreads();

  const int t = threadIdx.x;
  if (t < 16 * DEMB) {
    float s = accs[0][t] + accs[1][t] + accs[2][t] + accs[3][t];
    xe[(size_t)rowBase * DEMB + t] = s;            // coalesced 256B per block
  }
}

// ---------------------------------------------------------------------------
// K3: centroid (mean over N per dim) and scale = 0.9 / max|xe - centroid|.
// Single block, two-phase LDS reduction. stats = {c0,c1,c2,c3,scale}.
// ---------------------------------------------------------------------------
__global__ void k_stats(const float* __restrict__ xe, float* __restrict__ stats) {
  __shared__ float red[256 * 4];
  __shared__ float cent[4];
  const int t = threadIdx.x;
  float s0 = 0.f, s1 = 0.f, s2 = 0.f, s3 = 0.f;
  for (int r = t; r < NN; r += 256) {
    s0 += xe[r * 4 + 0]; s1 += xe[r * 4 + 1];
    s2 += xe[r * 4 + 2]; s3 += xe[r * 4 + 3];
  }
  red[t] = s0; red[256 + t] = s1; red[512 + t] = s2; red[768 + t] = s3;
  __syncthreads();
  for (int off = 128; off > 0; off >>= 1) {
    if (t < off) {
      red[t]       += red[t + off];
      red[256 + t] += red[256 + t + off];
      red[512 + t] += red[512 + t + off];
      red[768 + t] += red[768 + t + off];
    }
    __syncthreads();
  }
  if (t < 4) cent[t] = red[t * 256] / (float)NN;
  __syncthreads();
  const float c0 = cent[0], c1 = cent[1], c2 = cent[2], c3 = cent[3];
  float mx = 0.f;
  for (int r = t; r < NN; r += 256) {
    mx = fmaxf(mx, fabsf(xe[r * 4 + 0] - c0));
    mx = fmaxf(mx, fabsf(xe[r * 4 + 1] - c1));
    mx = fmaxf(mx, fabsf(xe[r * 4 + 2] - c2));
    mx = fmaxf(mx, fabsf(xe[r * 4 + 3] - c3));
  }
  red[t] = mx;
  __syncthreads();
  for (int off = 128; off > 0; off >>= 1) {
    if (t < off) red[t] = fmaxf(red[t], red[t + off]);
    __syncthreads();
  }
  if (t == 0) {
    stats[0] = c0; stats[1] = c1; stats[2] = c2; stats[3] = c3;
    stats[4] = 0.9f / red[0];
  }
}

// ---------------------------------------------------------------------------
// K4: z = (xe - centroid) * scale (b128 in/out), plus xe -> out_xe copy.
// ---------------------------------------------------------------------------
__global__ void k_z(const float* __restrict__ xe, const float* __restrict__ st,
                    float* __restrict__ z, float* __restrict__ out_xe) {
  int r = blockIdx.x * blockDim.x + threadIdx.x;
  if (r >= NN) return;
  float4 v = ((const float4*)xe)[r];
  ((float4*)out_xe)[r] = v;
  const float sc = st[4];
  float4 zz;
  zz.x = (v.x - st[0]) * sc;
  zz.y = (v.y - st[1]) * sc;
  zz.z = (v.z - st[2]) * sc;
  zz.w = (v.w - st[3]) * sc;
  ((float4*)z)[r] = zz;
}

// ---------------------------------------------------------------------------
// K5: per-row top-10 of Asim == top-10 of score s_j = 2*dot(z_i,z_j) - |z_j|^2
// (sigmoid monotone-decreasing in Dm = |z_i|^2 - s_j, clamped at 0).
// One wave per row; z_j fetched as one b128, |z_j|^2 computed inline.
// Zero backdrop written nontemporal+coalesced during the scan; the 10 sigmoid
// values scattered afterwards by lane 0 (same-wave stores are ordered).
// ---------------------------------------------------------------------------
__global__ void k_topk(const float* __restrict__ z,
                       const float* __restrict__ temp_p, const float* __restrict__ thr_p,
                       float* __restrict__ Aout) {
  __shared__ float sval[8][32 * TOPK];
  __shared__ int   sidx[8][32 * TOPK];
  const int lane = threadIdx.x & 31;
  const int wave = threadIdx.x >> 5;
  const int row  = blockIdx.x * 8 + wave;

  const float4 zi = ((const float4*)z)[row];
  const float sqi = zi.x * zi.x + zi.y * zi.y + zi.z * zi.z + zi.w * zi.w;

  float bv[TOPK];
  int   bi[TOPK];
#pragma unroll
  for (int t = 0; t < TOPK; ++t) { bv[t] = -3.0e38f; bi[t] = -1; }

  float* outRow = Aout + (size_t)row * NN;
  for (int j = lane; j < NN; j += 32) {
    float4 zj = ((const float4*)z)[j];
    float dot = zi.x * zj.x + zi.y * zj.y + zi.z * zj.z + zi.w * zj.w;
    float sqj = zj.x * zj.x + zj.y * zj.y + zj.z * zj.z + zj.w * zj.w;
    float s = 2.f * dot - sqj;
    __builtin_nontemporal_store(0.f, outRow + j);   // stream-once backdrop
    if (s > bv[TOPK - 1]) {
      float cv = s; int ci = j;
#pragma unroll
      for (int t = 0; t < TOPK; ++t) {
        if (cv > bv[t]) {
          float tv = bv[t]; int ti = bi[t];
          bv[t] = cv; bi[t] = ci; cv = tv; ci = ti;
        }
      }
    }
  }

#pragma unroll
  for (int t = 0; t < TOPK; ++t) {
    sval[wave][lane * TOPK + t] = bv[t];
    sidx[wave][lane * TOPK + t] = bi[t];
  }
  __syncthreads();

  if (lane == 0) {
    float mv[TOPK]; int mi[TOPK];
#pragma unroll
    for (int t = 0; t < TOPK; ++t) { mv[t] = -3.0e38f; mi[t] = -1; }
    for (int q = 0; q < 32 * TOPK; ++q) {
      float cv = sval[wave][q]; int ci = sidx[wave][q];
      if (cv > mv[TOPK - 1]) {
#pragma unroll
        for (int t = 0; t < TOPK; ++t) {
          if (cv > mv[t]) {
            float tv = mv[t]; int ti = mi[t];
            mv[t] = cv; mi[t] = ci; cv = tv; ci = ti;
          }
        }
      }
    }
    const float T   = temp_p[0];
    const float thr = fabsf(thr_p[0]);
    for (int t = 0; t < TOPK; ++t) {
      int j = mi[t];
      if (j < 0) continue;
      float4 zj = ((const float4*)z)[j];
      float dot = zi.x * zj.x + zi.y * zj.y + zi.z * zj.z + zi.w * zj.w;
      float sqj = zj.x * zj.x + zj.y * zj.y + zj.z * zj.z + zj.w * zj.w;
      float Dm  = fmaxf(sqi + sqj - 2.f * dot, 0.f);
      float a   = T * (thr - Dm);
      outRow[j] = 1.f / (1.f + expf(-a));
    }
  }
}

// ---------------------------------------------------------------------------
extern "C" void kernel_launch(void* const* d_in, const int* in_sizes, int n_in,
                              void* d_out, int out_size, void* d_ws, size_t ws_size,
                              hipStream_t stream) {
  const float* x    = (const float*)d_in[0];
  const float* A    = (const float*)d_in[1];
  const float* W    = (const float*)d_in[2];
  const float* temp = (const float*)d_in[3];
  const float* thr  = (const float*)d_in[4];

  float* out    = (float*)d_out;
  float* out_xe = out;                       // [8192*4]
  float* out_A  = out + (size_t)NN * DEMB;   // [8192*8192]

  char* ws = (char*)d_ws;
  float* Vt  = (float*)(ws);                 // 16*8192 f32 = 512 KB (transposed)
  float* xe  = (float*)(ws + 512 * 1024);    // 8192*4       128 KB
  float* z   = (float*)(ws + 640 * 1024);    // 8192*4       128 KB
  float* st  = (float*)(ws + 768 * 1024);    // 5 floats

  k_xw   <<<dim3((NN * 4) / 256), dim3(256), 0, stream>>>(x, W, Vt);
  k_gemm <<<dim3(NN / 16),        dim3(32 * KCHUNKS), 0, stream>>>(A, Vt, xe);
  k_stats<<<dim3(1),              dim3(256), 0, stream>>>(xe, st);
  k_z    <<<dim3(NN / 256),       dim3(256), 0, stream>>>(xe, st, z, out_xe);
  k_topk <<<dim3(NN / 8),         dim3(256), 0, stream>>>(z, temp, thr, out_A);
}